// ROIAlignRotated_23845658427952
// MI455X (gfx1250) — compile-verified
//
#include <hip/hip_runtime.h>

typedef __attribute__((ext_vector_type(2))) float v2f;
typedef __attribute__((ext_vector_type(8))) float v8f;

#define RA_C     256
#define RA_H     200
#define RA_W     304
#define RA_PH    7
#define RA_PW    7
#define RA_BINS  49      // 7*7
#define RA_NSAMP 196     // 49 bins * 4 samples (G=2 -> 2x2)
#define RA_PLANE (RA_H * RA_W)

// One block per ROI. 256 threads = 8 wave32.
// Per (bin, sample) one V_WMMA_F32_16X16X4_F32:
//   A(16x4) = gathered corner values for 16 channels (2 f32/lane:
//             lanes 0-15 hold corners K=0,1; lanes 16-31 hold K=2,3)
//   B(4x16) = bilinear weights in the single column N == bin-within-group,
//             pre-scaled by 0.25 (G*G average) and the validity mask
//   C/D     = f32 16x16 accumulator (channels x bins)
//
// LDS packs, per sample and per lane-half, 16 bytes: {offB0, offB1, w0, w1}
// -> inner loop: 1x ds_load_b128 + 2x global_load_b32 (saddr form) per
// sample, then 4 back-to-back WMMAs per bin. All loop control is pinned to
// SGPRs via readfirstlane so EXEC stays all-1s through the WMMA region.
__global__ __launch_bounds__(256) void roi_align_rotated_wmma(
    const float* __restrict__ inp,
    const float* __restrict__ rois,
    float* __restrict__ out)
{
    // [sid][half][0..1] = corner byte offsets, [2..3] = float weights (as bits)
    __shared__ alignas(16) int s_pk[RA_NSAMP][2][4];

    const int r   = blockIdx.x;
    const int tid = threadIdx.x;

    // ---------- per-ROI sampling geometry -> LDS (threads 0..195) ----------
    if (tid < RA_NSAMP) {
        const float scale = 0.25f;                       // SPATIAL_SCALE
        const float cxr = rois[r * 6 + 1] * scale - 0.5f;
        const float cyr = rois[r * 6 + 2] * scale - 0.5f;
        const float rw  = rois[r * 6 + 3] * scale;
        const float rh  = rois[r * 6 + 4] * scale;
        const float th  = rois[r * 6 + 5] * 0.017453292519943295f; // pi/180
        const float ct  = cosf(th);
        const float st  = sinf(th);
        const float bin_h = rh * (1.0f / RA_PH);
        const float bin_w = rw * (1.0f / RA_PW);

        const int bin = tid >> 2;          // 0..48
        const int s   = tid & 3;           // gy*2 + gx
        const int ph  = bin / RA_PW;
        const int pw  = bin % RA_PW;
        const int gy  = s >> 1;
        const int gx  = s & 1;

        const float yy = -0.5f * rh + (float)ph * bin_h + ((float)gy + 0.5f) * bin_h * 0.5f;
        const float xx = -0.5f * rw + (float)pw * bin_w + ((float)gx + 0.5f) * bin_w * 0.5f;
        const float y  = yy * ct - xx * st + cyr;
        const float x  = yy * st + xx * ct + cxr;

        const bool valid = (y >= -1.0f) && (y <= (float)RA_H) &&
                           (x >= -1.0f) && (x <= (float)RA_W);
        const float yc = fminf(fmaxf(y, 0.0f), (float)(RA_H - 1));
        const float xc = fminf(fmaxf(x, 0.0f), (float)(RA_W - 1));
        const int yl = (int)floorf(yc);
        const int xl = (int)floorf(xc);
        const int yh = min(yl + 1, RA_H - 1);
        const int xh = min(xl + 1, RA_W - 1);
        const float ly = yc - (float)yl, lx = xc - (float)xl;
        const float hy = 1.0f - ly,      hx = 1.0f - lx;
        const float vs = valid ? 0.25f : 0.0f;   // fold 1/(G*G) average + validity

        // corners 0,1 (top row) for lane-half 0; corners 2,3 (bottom) for half 1
        s_pk[tid][0][0] = (yl * RA_W + xl) * 4;
        s_pk[tid][0][1] = (yl * RA_W + xh) * 4;
        s_pk[tid][0][2] = __float_as_int(hy * hx * vs);
        s_pk[tid][0][3] = __float_as_int(hy * lx * vs);
        s_pk[tid][1][0] = (yh * RA_W + xl) * 4;
        s_pk[tid][1][1] = (yh * RA_W + xh) * 4;
        s_pk[tid][1][2] = __float_as_int(ly * hx * vs);
        s_pk[tid][1][3] = __float_as_int(ly * lx * vs);
    }
    __syncthreads();

    // Wave id is uniform within a wave: pin to SGPR so all loop control below
    // is scalar (s_cmp/s_cbranch) and EXEC stays all-1s for the WMMAs.
    const int wave = __builtin_amdgcn_readfirstlane(tid >> 5);
    const int lane = tid & 31;
    const int m    = lane & 15;        // A row (channel-in-group) == B col index
    const int hsel = lane >> 4;        // 0 -> corners {0,1}, 1 -> corners {2,3}

    // batch index is uniform per block: pin the plane base into SGPRs
    const int b = __builtin_amdgcn_readfirstlane((int)rois[r * 6 + 0]);
    const char* base  = (const char*)(inp + (size_t)b * (RA_C * RA_PLANE));
    float*      obase = out + (size_t)r * (RA_C * RA_BINS);

    // 64 tiles: cg = channel group (16 ch), bg = bin group (16 bins).
    // cg=q&15, bg=q>>4 balances the short bg==3 group across the 8 waves.
    for (int q = wave; q < 64; q += 8) {
        const int cg = q & 15;           // scalar
        const int bg = q >> 4;           // scalar
        const unsigned planeB = (unsigned)((cg * 16 + m) * RA_PLANE) * 4u;

        v8f acc = {0.f, 0.f, 0.f, 0.f, 0.f, 0.f, 0.f, 0.f};
        const int nb = (bg == 3) ? (RA_BINS - 48) : 16;   // scalar trip count

        for (int bp = 0; bp < nb; ++bp) {
            const int   bin  = bg * 16 + bp;              // scalar
            const float hitf = (m == bp) ? 1.0f : 0.0f;   // owns B column bp

            v2f av[4], wv[4];
#pragma unroll
            for (int s = 0; s < 4; ++s) {
                const int4 pk = *(const int4*)&s_pk[bin * 4 + s][hsel][0];
                av[s].x = *(const float*)(base + (planeB + (unsigned)pk.x));
                av[s].y = *(const float*)(base + (planeB + (unsigned)pk.y));
                wv[s].x = __int_as_float(pk.z) * hitf;
                wv[s].y = __int_as_float(pk.w) * hitf;
            }
#pragma unroll
            for (int s = 0; s < 4; ++s) {
                acc = __builtin_amdgcn_wmma_f32_16x16x4_f32(
                    /*neg_a=*/false, av[s], /*neg_b=*/false, wv[s],
                    /*c_mod=*/(short)0, acc,
                    /*reuse_a=*/false, /*reuse_b=*/false);
            }
        }

        // D layout: VGPR i -> M = i (lanes 0-15) or i+8 (lanes 16-31); N = lane&15
        const int Mbase = (lane < 16) ? 0 : 8;
        const int bin   = bg * 16 + m;     // N index == m
        if (bin < RA_BINS) {
            float* op = obase + (cg * 16 + Mbase) * RA_BINS + bin;
#pragma unroll
            for (int i = 0; i < 8; ++i)
                op[i * RA_BINS] = acc[i];
        }
    }
}

extern "C" void kernel_launch(void* const* d_in, const int* in_sizes, int n_in,
                              void* d_out, int out_size, void* d_ws, size_t ws_size,
                              hipStream_t stream) {
    const float* inp  = (const float*)d_in[0];
    const float* rois = (const float*)d_in[1];
    float* out        = (float*)d_out;
    const int R = in_sizes[1] / 6;   // 1000
    roi_align_rotated_wmma<<<R, 256, 0, stream>>>(inp, rois, out);
}